// TAGConv_3l_128h_w_k5_52896817218190
// MI455X (gfx1250) — compile-verified
//
#include <hip/hip_runtime.h>
#include <math.h>

// Problem constants (fixed by the reference)
#define F_IN  100
#define HID   128
#define C_OUT 10
#define KHOPS 5

typedef __attribute__((ext_vector_type(2))) float v2f;
typedef __attribute__((ext_vector_type(8))) float v8f;

// D = A(16x4 f32) * B(4x16 f32) + C(16x16 f32), full-precision matrix pipe.
__device__ __forceinline__ v8f wmma_f32x4(v2f a, v2f b, v8f c) {
  return __builtin_amdgcn_wmma_f32_16x16x4_f32(
      /*neg_a=*/false, a, /*neg_b=*/false, b,
      /*c_mod=*/(short)0, c, /*reuse_a=*/false, /*reuse_b=*/false);
}

__global__ void zero_f32_kernel(float* __restrict__ p, int n) {
  int i = blockIdx.x * blockDim.x + threadIdx.x;
  if (i < n) p[i] = 0.0f;
}

// Transpose + zero-pad hop weights:
//   W  : (K+1) x d x hout (row-major) ->  WT : (K+1) x n16 x d (col-contiguous)
__global__ void wt_kernel(const float* __restrict__ W, float* __restrict__ WT,
                          int d, int hout, int n16, int total) {
  int i = blockIdx.x * blockDim.x + threadIdx.x;
  if (i >= total) return;
  int k  = i / (n16 * d);
  int r  = i - k * (n16 * d);
  int n  = r / d;
  int kd = r - n * d;
  WT[i] = (n < hout) ? W[((size_t)k * d + kd) * hout + n] : 0.0f;
}

// deg[col[e]] += w[e]   (hardware f32 atomic, L2-resident)
__global__ void degree_kernel(const long long* __restrict__ col,
                              const float* __restrict__ w,
                              float* __restrict__ deg, int e) {
  int i = blockIdx.x * blockDim.x + threadIdx.x;
  if (i < e) unsafeAtomicAdd(&deg[(int)col[i]], w[i]);
}

// deg -> D^{-1/2} in place
__global__ void invsqrt_kernel(float* __restrict__ deg, int n) {
  int i = blockIdx.x * blockDim.x + threadIdx.x;
  if (i < n) {
    float d = deg[i];
    deg[i] = (d > 0.0f) ? rsqrtf(fmaxf(d, 1e-12f)) : 0.0f;
  }
}

// norm[e] = dis[row]*w*dis[col]
__global__ void norm_kernel(const long long* __restrict__ row,
                            const long long* __restrict__ col,
                            const float* __restrict__ w,
                            const float* __restrict__ dis,
                            float* __restrict__ nrm, int e) {
  int i = blockIdx.x * blockDim.x + threadIdx.x;
  if (i < e) nrm[i] = dis[(int)row[i]] * w[i] * dis[(int)col[i]];
}

// ---- CSC build (once per launch) ----
__global__ void count_kernel(const long long* __restrict__ col, int* __restrict__ cnt, int e) {
  int i = blockIdx.x * blockDim.x + threadIdx.x;
  if (i < e) atomicAdd(&cnt[(int)col[i]], 1);
}

// Single-block exclusive scan (LDS Hillis-Steele over 1024-chunks, running carry)
__global__ void scan_kernel(const int* __restrict__ cnt, int* __restrict__ rowptr, int n) {
  __shared__ int buf[1024];
  __shared__ int carry;
  if (threadIdx.x == 0) carry = 0;
  __syncthreads();
  for (int base = 0; base < n; base += 1024) {
    int i = base + threadIdx.x;
    int v = (i < n) ? cnt[i] : 0;
    buf[threadIdx.x] = v;
    __syncthreads();
    for (int off = 1; off < 1024; off <<= 1) {
      int t = (threadIdx.x >= (unsigned)off) ? buf[threadIdx.x - off] : 0;
      __syncthreads();
      buf[threadIdx.x] += t;
      __syncthreads();
    }
    if (i < n) rowptr[i] = carry + buf[threadIdx.x] - v;   // exclusive
    __syncthreads();                                       // all carry reads done
    if (threadIdx.x == 1023) carry += buf[1023];
    __syncthreads();
  }
  if (threadIdx.x == 0) rowptr[n] = carry;
}

// Scatter packed edge records (src row, norm) into CSC order
__global__ void fill_kernel(const long long* __restrict__ row,
                            const long long* __restrict__ col,
                            const float* __restrict__ nrm,
                            const int* __restrict__ rowptr,
                            int* __restrict__ cursor,
                            int2* __restrict__ er, int e) {
  int i = blockIdx.x * blockDim.x + threadIdx.x;
  if (i < e) {
    int c = (int)col[i];
    int pos = rowptr[c] + atomicAdd(&cursor[c], 1);
    int2 v;
    v.x = (int)row[i];
    v.y = __float_as_int(nrm[i]);
    er[pos] = v;
  }
}

// Gather-reduce hop: h_out[c,:] = sum_{e in in(c)} norm_e * h_in[row_e,:]
// One wave per destination node; lane covers 4 features. No atomics, no zero pass.
__global__ void __launch_bounds__(256)
gather_kernel(const int2* __restrict__ er, const int* __restrict__ rowptr,
              const float* __restrict__ hin, float* __restrict__ hout,
              int d, int n) {
  const int node = blockIdx.x * 8 + (threadIdx.x >> 5);
  if (node >= n) return;
  const int lane = threadIdx.x & 31;
  int f = lane * 4;
  const bool act = (f < d);
  if (f > d - 4) f = d - 4;            // clamp idle lanes (d=100): no divergence in loop
  const int p0 = rowptr[node], p1 = rowptr[node + 1];   // wave-uniform
  float4 acc = make_float4(0.f, 0.f, 0.f, 0.f);
  for (int p = p0; p < p1; ++p) {
    int2 rw = er[p];                   // one b64 per edge, wave-broadcast
    float w = __int_as_float(rw.y);
    const float4 v = *(const float4*)(hin + (size_t)rw.x * d + f);
    acc.x += w * v.x; acc.y += w * v.y; acc.z += w * v.z; acc.w += w * v.w;
  }
  if (act) *(float4*)(hout + (size_t)node * d + f) = acc;
}

// out[N x hout] (+)= h[N x d] @ WT (+ bias on init). WT padded/transposed [n16 x d].
// f32 WMMA lane layout (ISA 7.12.2): A 16x4: lanes0-15 rows M, v0/v1 = K,K+1;
// lanes16-31 same rows, K+2,K+3. B mirrored. C/D: VGPR i = row m0+i (+8 hi half).
__global__ void __launch_bounds__(256)
gemm_tag_kernel(const float* __restrict__ h,
                const float* __restrict__ WT,
                const float* __restrict__ bias,
                float* __restrict__ out,
                int nrows, int d, int hout, int init) {
  const int m0   = blockIdx.x * 16;
  const int wave = threadIdx.x >> 5;
  const int lane = threadIdx.x & 31;
  const int nl   = lane & 15;
  const int hi   = lane >> 4;
  const int cidx = wave * 16 + nl;     // padded column index (always valid in WT)
  const bool colok = (cidx < hout);

  v8f acc;
  if (init) {
    float bv = colok ? bias[cidx] : 0.0f;
    #pragma unroll
    for (int i = 0; i < 8; ++i) acc[i] = bv;
  } else {
    #pragma unroll
    for (int i = 0; i < 8; ++i) {
      int r = m0 + i + hi * 8;
      acc[i] = (colok && r < nrows) ? out[(size_t)r * hout + cidx] : 0.0f;
    }
  }

  int arow = m0 + nl;
  if (arow >= nrows) arow = nrows - 1;
  const float* hrow = h + (size_t)arow * d + hi * 2;
  const float* wcol = WT + (size_t)cidx * d + hi * 2;

  for (int kk = 0; kk < d; kk += 4) {  // d multiple of 4 (100 or 128)
    v2f a = *(const v2f*)(hrow + kk);  // global_load_b64, unconditional
    v2f b = *(const v2f*)(wcol + kk);  // global_load_b64, unconditional
    acc = wmma_f32x4(a, b, acc);       // EXEC all-ones: uniform flow here
  }

  #pragma unroll
  for (int i = 0; i < 8; ++i) {
    int r = m0 + i + hi * 8;
    if (colok && r < nrows) out[(size_t)r * hout + cidx] = acc[i];
  }
}

// ELU (alpha = 1)
__global__ void elu_kernel(const float* __restrict__ in, float* __restrict__ out, int n) {
  int i = blockIdx.x * blockDim.x + threadIdx.x;
  if (i < n) {
    float v = in[i];
    out[i] = (v > 0.0f) ? v : (expf(v) - 1.0f);
  }
}

static inline int blks(int n, int t) { return (n + t - 1) / t; }

extern "C" void kernel_launch(void* const* d_in, const int* in_sizes, int n_in,
                              void* d_out, int out_size, void* d_ws, size_t ws_size,
                              hipStream_t stream) {
  const float*     x   = (const float*)d_in[0];
  const long long* ei  = (const long long*)d_in[1];   // int64 [2, E]
  const float*     ew  = (const float*)d_in[2];
  const float*     W1  = (const float*)d_in[3];       // (6, 100, 128)
  const float*     b1  = (const float*)d_in[4];
  const float*     W2  = (const float*)d_in[5];       // (6, 128, 128)
  const float*     b2  = (const float*)d_in[6];
  const float*     W3  = (const float*)d_in[7];       // (6, 128, 10)
  const float*     b3  = (const float*)d_in[8];
  float*           out = (float*)d_out;

  const int E = in_sizes[2];
  const int N = in_sizes[0] / F_IN;
  const long long* row = ei;
  const long long* col = ei + E;

  // Workspace carve-out (~88 MB, L2-resident on MI455X)
  char*  ws     = (char*)d_ws;
  float* nrm    = (float*)ws;                       // E
  float* deg    = nrm + E;                          // N  (becomes D^{-1/2})
  float* wt1    = deg + N;                          // 6*128*100
  float* wt2    = wt1 + (KHOPS + 1) * HID * F_IN;   // 6*128*128
  float* wt3    = wt2 + (KHOPS + 1) * HID * HID;    // 6*16*128
  int*   cnt    = (int*)(wt3 + (KHOPS + 1) * 16 * HID);  // N (counts then cursor)
  int*   rowptr = cnt + N;                          // N+1
  int2*  er     = (int2*)(((size_t)(rowptr + N + 1) + 255) & ~(size_t)255); // E
  float* bufA   = (float*)(((size_t)(er + E) + 255) & ~(size_t)255);        // N*HID
  float* bufB   = bufA + (size_t)N * HID;
  float* bufO   = bufB + (size_t)N * HID;

  const int T = 256;
  const int mtiles = (N + 15) / 16;                 // 3125 exactly
  const int gblk   = (N + 7) / 8;                   // gather: 8 waves/block

  // ---- GCN normalization ----
  zero_f32_kernel<<<blks(N, T), T, 0, stream>>>(deg, N);
  degree_kernel<<<blks(E, T), T, 0, stream>>>(col, ew, deg, E);
  invsqrt_kernel<<<blks(N, T), T, 0, stream>>>(deg, N);
  norm_kernel<<<blks(E, T), T, 0, stream>>>(row, col, ew, deg, nrm, E);

  // ---- CSC build (once; removes 1.28e9 scatter atomics from the 15 hops) ----
  zero_f32_kernel<<<blks(N, T), T, 0, stream>>>((float*)cnt, N);
  count_kernel<<<blks(E, T), T, 0, stream>>>(col, cnt, E);
  scan_kernel<<<1, 1024, 0, stream>>>(cnt, rowptr, N);
  zero_f32_kernel<<<blks(N, T), T, 0, stream>>>((float*)cnt, N);   // reuse as cursor
  fill_kernel<<<blks(E, T), T, 0, stream>>>(row, col, nrm, rowptr, cnt, er, E);

  // ---- Weight transpose/pad (tiny, once per launch) ----
  {
    int t1 = (KHOPS + 1) * HID * F_IN;
    int t2 = (KHOPS + 1) * HID * HID;
    int t3 = (KHOPS + 1) * 16 * HID;
    wt_kernel<<<blks(t1, T), T, 0, stream>>>(W1, wt1, F_IN, HID,   HID, t1);
    wt_kernel<<<blks(t2, T), T, 0, stream>>>(W2, wt2, HID,  HID,   HID, t2);
    wt_kernel<<<blks(t3, T), T, 0, stream>>>(W3, wt3, HID,  C_OUT, 16,  t3);
  }

  // ---- Layer 1: F_IN=100 -> HID=128 ----
  gemm_tag_kernel<<<mtiles, 8 * 32, 0, stream>>>(x, wt1, b1, bufO, N, F_IN, HID, 1);
  {
    const float* hcur = x;
    float* ping = bufA; float* pong = bufB;
    for (int k = 1; k <= KHOPS; ++k) {
      float* hnext = ping;
      gather_kernel<<<gblk, T, 0, stream>>>(er, rowptr, hcur, hnext, F_IN, N);
      gemm_tag_kernel<<<mtiles, 8 * 32, 0, stream>>>(hnext, wt1 + (size_t)k * HID * F_IN,
                                                     b1, bufO, N, F_IN, HID, 0);
      hcur = hnext; float* t2 = ping; ping = pong; pong = t2;
    }
  }
  elu_kernel<<<blks(N * HID, T), T, 0, stream>>>(bufO, bufA, N * HID);

  // ---- Layer 2: HID -> HID ----
  gemm_tag_kernel<<<mtiles, 8 * 32, 0, stream>>>(bufA, wt2, b2, bufO, N, HID, HID, 1);
  {
    const float* hcur = bufA;
    float* ping = bufB; float* pong = bufA;
    for (int k = 1; k <= KHOPS; ++k) {
      float* hnext = ping;
      gather_kernel<<<gblk, T, 0, stream>>>(er, rowptr, hcur, hnext, HID, N);
      gemm_tag_kernel<<<mtiles, 8 * 32, 0, stream>>>(hnext, wt2 + (size_t)k * HID * HID,
                                                     b2, bufO, N, HID, HID, 0);
      hcur = hnext; float* t2 = ping; ping = pong; pong = t2;
    }
  }
  elu_kernel<<<blks(N * HID, T), T, 0, stream>>>(bufO, bufA, N * HID);

  // ---- Layer 3: HID -> C_OUT (writes d_out; 1 wave/block, cols padded to 16) ----
  gemm_tag_kernel<<<mtiles, 1 * 32, 0, stream>>>(bufA, wt3, b3, out, N, HID, C_OUT, 1);
  {
    const float* hcur = bufA;
    float* ping = bufB; float* pong = bufA;
    for (int k = 1; k <= KHOPS; ++k) {
      float* hnext = ping;
      gather_kernel<<<gblk, T, 0, stream>>>(er, rowptr, hcur, hnext, HID, N);
      gemm_tag_kernel<<<mtiles, 1 * 32, 0, stream>>>(hnext, wt3 + (size_t)k * 16 * HID,
                                                     b3, out, N, HID, C_OUT, 0);
      hcur = hnext; float* t2 = ping; ping = pong; pong = t2;
    }
  }
}